// SeparateLoss_46385646796817
// MI455X (gfx1250) — compile-verified
//
#include <hip/hip_runtime.h>
#include <hip/hip_bf16.h>

// SeparateLoss on MI455X (gfx1250, wave32).
//
// Algebra: with binary labels, mean(mask * (Fn @ Fn^T)) == 2 * dot(S0, S1) / N^2
// where S_c = sum of L2-normalized rows with label c. This turns a 34-GFLOP,
// 1-GB-traffic N x N Gram problem into a 4-8 MB bandwidth-bound reduction.
// The masked accumulation S = M(2xN one-hot) @ Fn(Nx64) is executed on the
// matrix pipe with V_WMMA_F32_16X16X4_F32 (4 rows x 64 channels per wave-iter,
// 4 channel tiles / accumulators).

typedef __attribute__((ext_vector_type(2))) float v2f;
typedef __attribute__((ext_vector_type(8))) float v8f;

#define NROWS    16384     // 4 * 64 * 64
#define NGROUPS  4096      // 4 rows per WMMA group
#define NBLOCKS  64
#define TPB      256
#define WPB      8         // waves per block (wave32)
#define TOTWAVES (NBLOCKS * WPB)   // 512 -> 8 uniform iterations per wave

__global__ __launch_bounds__(TPB)
void separate_loss_partials(const float* __restrict__ feat,
                            const int*   __restrict__ label,
                            float*       __restrict__ ws) {
    __shared__ float sm[WPB][128];

    const int lane = threadIdx.x & 31;
    const int wib  = threadIdx.x >> 5;            // wave in block
    const int wid  = blockIdx.x * WPB + wib;      // global wave id (uniform)
    const int half = lane >> 4;                   // 0: K rows {0,1}, 1: K rows {2,3}
    const int c    = lane & 15;                   // M index for A, N (channel) for B

    v8f acc[4];
    #pragma unroll
    for (int t = 0; t < 4; ++t) acc[t] = (v8f){};

    // Uniform trip count (4096 % 512 == 0) -> EXEC all-ones at every WMMA.
    for (int g = wid; g < NGROUPS; g += TOTWAVES) {
        // This half-wave owns rows ra (K = 2*half) and rb = ra+1 (K = 2*half+1).
        const int ra = 4 * g + 2 * half;
        const int rb = ra + 1;

        const int ba = ra >> 12, rema = ra & 4095;
        const int ya = rema >> 6, xa = rema & 63;
        const int bb = rb >> 12, remb = rb & 4095;
        const int yb = remb >> 6, xb = remb & 63;

        // nearest-resize source pixel (2y, 2x) in the 128x128 plane
        const int pixa = ya * 256 + xa * 2;
        const int pixb = yb * 256 + xb * 2;
        const float* fa = feat + (size_t)ba * 1048576 + pixa;
        const float* fb = feat + (size_t)bb * 1048576 + pixb;
        const int laba = label[ba * 16384 + pixa];
        const int labb = label[bb * 16384 + pixb];

        // Gather 64 channels of both rows across the 16 lanes of this half.
        float f0[4], f1[4];
        #pragma unroll
        for (int t = 0; t < 4; ++t) {
            const int ch = 16 * t + c;
            f0[t] = fa[(size_t)ch * 16384];
            f1[t] = fb[(size_t)ch * 16384];
        }

        // Row L2 norms: per-lane partial, then 16-lane butterfly (stays in half).
        float ss0 = f0[0]*f0[0] + f0[1]*f0[1] + f0[2]*f0[2] + f0[3]*f0[3];
        float ss1 = f1[0]*f1[0] + f1[1]*f1[1] + f1[2]*f1[2] + f1[3]*f1[3];
        #pragma unroll
        for (int m = 1; m < 16; m <<= 1) {
            ss0 += __shfl_xor(ss0, m, 32);
            ss1 += __shfl_xor(ss1, m, 32);
        }
        const float inv0 = 1.0f / fmaxf(sqrtf(ss0), 1e-12f);
        const float inv1 = 1.0f / fmaxf(sqrtf(ss1), 1e-12f);

        // A (16x4 f32): lane L<16 -> M=L, VGPR0=K0, VGPR1=K1; lanes 16-31 -> K2/K3.
        // A[m][k] = (label(row k) == m); rows >= 2 never match (labels are 0/1).
        v2f A;
        A.x = (laba == c) ? 1.0f : 0.0f;
        A.y = (labb == c) ? 1.0f : 0.0f;

        // B (4x16 f32) per channel tile: lanes hold N=c; VGPR0=K0/K2, VGPR1=K1/K3.
        v2f B[4];
        #pragma unroll
        for (int t = 0; t < 4; ++t) {
            B[t].x = f0[t] * inv0;
            B[t].y = f1[t] * inv1;
        }

        #pragma unroll
        for (int t = 0; t < 4; ++t) {
            acc[t] = __builtin_amdgcn_wmma_f32_16x16x4_f32(
                /*neg_a=*/false, A, /*neg_b=*/false, B[t],
                /*c_mod=*/(short)0, acc[t], /*reuse_a=*/false, /*reuse_b=*/false);
        }
    }

    // D layout: VGPR0 = row M=0 (S0) on lanes 0-15, VGPR1 = row M=1 (S1).
    if (lane < 16) {
        #pragma unroll
        for (int t = 0; t < 4; ++t) {
            sm[wib][16 * t + c]      = acc[t][0];
            sm[wib][64 + 16 * t + c] = acc[t][1];
        }
    }
    __syncthreads();

    // Deterministic block reduction (fixed order, no atomics), one slot per block.
    if (threadIdx.x < 128) {
        float s = 0.0f;
        #pragma unroll
        for (int w = 0; w < WPB; ++w) s += sm[w][threadIdx.x];
        ws[blockIdx.x * 128 + threadIdx.x] = s;
    }
}

__global__ __launch_bounds__(128)
void separate_loss_finalize(const float* __restrict__ ws,
                            float* __restrict__ out) {
    __shared__ float S[128];
    const int t = threadIdx.x;
    float s = 0.0f;
    for (int b = 0; b < NBLOCKS; ++b) s += ws[b * 128 + t];
    S[t] = s;
    __syncthreads();
    if (t == 0) {
        float dot = 0.0f;
        for (int i = 0; i < 64; ++i) dot += S[i] * S[i + 64];
        out[0] = 2.0f * dot / ((float)NROWS * (float)NROWS);
    }
}

extern "C" void kernel_launch(void* const* d_in, const int* in_sizes, int n_in,
                              void* d_out, int out_size, void* d_ws, size_t ws_size,
                              hipStream_t stream) {
    (void)in_sizes; (void)n_in; (void)out_size; (void)ws_size;
    const float* feat  = (const float*)d_in[0];
    const int*   label = (const int*)d_in[1];
    float*       ws    = (float*)d_ws;     // 64 blocks * 128 floats = 32 KB
    float*       out   = (float*)d_out;

    separate_loss_partials<<<NBLOCKS, TPB, 0, stream>>>(feat, label, ws);
    separate_loss_finalize<<<1, 128, 0, stream>>>(ws, out);
}